// AttentionPooler_75488345195011
// MI455X (gfx1250) — compile-verified
//
#include <hip/hip_runtime.h>
#include <hip/hip_bf16.h>
#include <math.h>

typedef __attribute__((ext_vector_type(2))) float v2f;
typedef __attribute__((ext_vector_type(8))) float v8f;

namespace {
constexpr int kLayer  = 2;
constexpr int kN = 32, kL = 512, kD = 1024, kH = 256;
constexpr int kS = 2048, kSpan = 64;
constexpr int kKC = 64;          // K-chunk staged in LDS
constexpr int kAStride = 68;     // padded A row stride (floats): conflict-free frag reads
constexpr int kWPairStride = 272;// k-pair stride (cols+pad): 2*272 % 64 == 32 -> halves disjoint
}

__launch_bounds__(256, 2)
__global__ void attention_pooler_kernel(const float* __restrict__ hs_all,
                                        const int*  __restrict__ spans,
                                        const float* __restrict__ Wm,
                                        const float* __restrict__ bvec,
                                        const float* __restrict__ vvec,
                                        float* __restrict__ out)
{
    // A: 64 rows x 64 k (padded).  Wp: pair-major W slab: [kp][col][2] floats,
    // so a B fragment {W[kb+2hi][col], W[kb+2hi+1][col]} is one aligned b64 load.
    __shared__ __align__(16) float As[kSpan * kAStride];                 // 17408 B
    __shared__ __align__(16) float Wp[(kKC / 2) * kWPairStride * 2];     // 69632 B
    __shared__ float sc[kSpan];
    __shared__ float att[kSpan];
    __shared__ float pooled[kH];
    __shared__ float red[2];

    const int s    = blockIdx.x;
    const int tid  = threadIdx.x;
    const int wave = tid >> 5;     // 8 waves (wave32)
    const int lane = tid & 31;
    const int mt   = wave & 3;     // M tile 0..3 (rows mt*16..mt*16+15)
    const int nh   = wave >> 2;    // N half: cols nh*128..nh*128+127
    const int hi   = lane >> 4;    // half-wave (K pair select in frags)
    const int l16  = lane & 15;

    const float* hs = hs_all + (size_t)kLayer * kN * kL * kD;

    const int idx   = spans[3 * s + 0];
    const int start = spans[3 * s + 1];
    const int stop  = spans[3 * s + 2];

    if (tid < kSpan) sc[tid]     = 0.f;
    if (tid < kH)    pooled[tid] = 0.f;

    v8f acc[8];
    #pragma unroll
    for (int nt = 0; nt < 8; ++nt)
        #pragma unroll
        for (int e = 0; e < 8; ++e) acc[nt][e] = 0.f;

    // A-gather assignment: 4 threads per span row, 16 floats each per chunk
    const int rowj = tid >> 2;
    const int quad = tid & 3;
    int pos = start + rowj;
    if (pos > kL - 1) pos = kL - 1;
    const float* arow = hs + ((size_t)idx * kL + pos) * kD;

    // W staging assignment: task t covers k-pair kp, cols 4*c4..4*c4+3
    for (int kc = 0; kc < kD / kKC; ++kc) {
        const int kbase = kc * kKC;
        __syncthreads();   // protect LDS reuse (also orders the zero-init above)

        // ---- stage gathered A rows: 64 x 64 fp32 ----
        {
            const float4* src = (const float4*)(arow + kbase + quad * 16);
            float4 t0 = src[0], t1 = src[1], t2 = src[2], t3 = src[3];
            float* dst = &As[rowj * kAStride + quad * 16];
            *(float4*)(dst +  0) = t0;
            *(float4*)(dst +  4) = t1;
            *(float4*)(dst +  8) = t2;
            *(float4*)(dst + 12) = t3;
        }
        // ---- stage W slab pair-major: rows kbase..kbase+63 x 256 cols ----
        // 8 tasks/thread, batched by 2 so global loads overlap.
        #pragma unroll
        for (int i0 = 0; i0 < 8; i0 += 2) {
            const int ta = tid + i0 * 256;
            const int tb = tid + (i0 + 1) * 256;
            const int kpa = ta >> 6, ca = (ta & 63) * 4;
            const int kpb = tb >> 6, cb = (tb & 63) * 4;
            const float* ga0 = Wm + (size_t)(kbase + 2 * kpa + 0) * kH + ca;
            const float* ga1 = Wm + (size_t)(kbase + 2 * kpa + 1) * kH + ca;
            const float* gb0 = Wm + (size_t)(kbase + 2 * kpb + 0) * kH + cb;
            const float* gb1 = Wm + (size_t)(kbase + 2 * kpb + 1) * kH + cb;
            float4 a0 = *(const float4*)ga0;
            float4 a1 = *(const float4*)ga1;
            float4 b0 = *(const float4*)gb0;
            float4 b1 = *(const float4*)gb1;
            float* da = &Wp[(kpa * kWPairStride + ca) * 2];
            float* db = &Wp[(kpb * kWPairStride + cb) * 2];
            v2f p;
            p.x = a0.x; p.y = a1.x; *(v2f*)(da + 0) = p;
            p.x = a0.y; p.y = a1.y; *(v2f*)(da + 2) = p;
            p.x = a0.z; p.y = a1.z; *(v2f*)(da + 4) = p;
            p.x = a0.w; p.y = a1.w; *(v2f*)(da + 6) = p;
            p.x = b0.x; p.y = b1.x; *(v2f*)(db + 0) = p;
            p.x = b0.y; p.y = b1.y; *(v2f*)(db + 2) = p;
            p.x = b0.z; p.y = b1.z; *(v2f*)(db + 4) = p;
            p.x = b0.w; p.y = b1.w; *(v2f*)(db + 6) = p;
        }
        __syncthreads();

        // ---- 16 k-steps of V_WMMA_F32_16X16X4_F32 ----
        #pragma unroll
        for (int kk = 0; kk < kKC / 4; ++kk) {
            const int kb = kk * 4;
            // A frag 16x4: lanes0-15 K{kb,kb+1}; lanes16-31 K{kb+2,kb+3} (one b64)
            const v2f a = *(const v2f*)&As[(mt * 16 + l16) * kAStride + kb + 2 * hi];
            // All 8 B frags first (one b64 each), then a dense WMMA burst.
            v2f bfr[8];
            #pragma unroll
            for (int nt = 0; nt < 8; ++nt) {
                const int col = (nh * 8 + nt) * 16 + l16;
                bfr[nt] = *(const v2f*)&Wp[((kb >> 1) + hi) * (kWPairStride * 2) + col * 2];
            }
            #pragma unroll
            for (int nt = 0; nt < 8; ++nt) {
                acc[nt] = __builtin_amdgcn_wmma_f32_16x16x4_f32(
                    false, a, false, bfr[nt], (short)0, acc[nt], false, false);
            }
        }
    }

    // ---- bias add + attention scores: srow[r] = sum_cols proj[row][col]*v[col] ----
    float srow[8];
    #pragma unroll
    for (int r = 0; r < 8; ++r) srow[r] = 0.f;
    #pragma unroll
    for (int nt = 0; nt < 8; ++nt) {
        const int col    = (nh * 8 + nt) * 16 + l16;
        const float bcol = bvec[col];
        const float vcol = vvec[col];
        #pragma unroll
        for (int r = 0; r < 8; ++r) {
            acc[nt][r] += bcol;              // proj = A@W + b (kept in regs)
            srow[r] += acc[nt][r] * vcol;
        }
    }
    // butterfly-reduce over the 16 lanes sharing the same rows
    #pragma unroll
    for (int off = 8; off >= 1; off >>= 1)
        #pragma unroll
        for (int r = 0; r < 8; ++r)
            srow[r] += __shfl_xor(srow[r], off, 16);
    if (l16 == 0) {
        #pragma unroll
        for (int r = 0; r < 8; ++r)
            atomicAdd(&sc[mt * 16 + hi * 8 + r], srow[r]);  // C/D: row = 8*hi + r
    }
    __syncthreads();

    // ---- masked softmax over the 64 positions ----
    const int len = stop - start;   // mask: l < len
    if (tid == 0) {
        float m = -INFINITY;
        for (int l = 0; l < kSpan; ++l)
            if (l < len) m = fmaxf(m, sc[l]);
        red[0] = m;
    }
    __syncthreads();
    if (tid < kSpan)
        att[tid] = (tid < len) ? expf(sc[tid] - red[0]) : 0.f;
    __syncthreads();
    if (tid == 0) {
        float sum = 0.f;
        for (int l = 0; l < kSpan; ++l) sum += att[l];
        red[1] = 1.f / sum;
    }
    __syncthreads();

    // ---- pooled[h] = sum_l att_norm[l] * proj[l][h] ----
    float aw[8];
    const float inv = red[1];
    #pragma unroll
    for (int r = 0; r < 8; ++r)
        aw[r] = att[mt * 16 + hi * 8 + r] * inv;
    #pragma unroll
    for (int nt = 0; nt < 8; ++nt) {
        float p = 0.f;
        #pragma unroll
        for (int r = 0; r < 8; ++r) p += aw[r] * acc[nt][r];
        p += __shfl_xor(p, 16, 32);          // fold the two half-wave row groups
        if (hi == 0) atomicAdd(&pooled[(nh * 8 + nt) * 16 + l16], p);
    }
    __syncthreads();

    out[(size_t)s * kH + tid] = pooled[tid]; // 256 threads == H, coalesced
}

extern "C" void kernel_launch(void* const* d_in, const int* in_sizes, int n_in,
                              void* d_out, int out_size, void* d_ws, size_t ws_size,
                              hipStream_t stream) {
    (void)in_sizes; (void)n_in; (void)out_size; (void)d_ws; (void)ws_size;
    const float* hs    = (const float*)d_in[0];
    const int*   spans = (const int*)  d_in[1];   // (S,3) int32: idx, start, stop
    const float* W     = (const float*)d_in[2];
    const float* b     = (const float*)d_in[3];
    const float* v     = (const float*)d_in[4];
    float* out = (float*)d_out;
    attention_pooler_kernel<<<dim3(kS), dim3(256), 0, stream>>>(hs, spans, W, b, v, out);
}